// GAT_31714038514427
// MI455X (gfx1250) — compile-verified
//
#include <hip/hip_runtime.h>
#include <hip/hip_bf16.h>

typedef float v2f __attribute__((ext_vector_type(2)));
typedef float v8f __attribute__((ext_vector_type(8)));

#define NEG_SLOPE 0.2f

// Order-preserving int encoding for float atomic max (NaN-free data).
__device__ __forceinline__ int fenc(float f) {
  int i = __float_as_int(f);
  return i >= 0 ? i : (i ^ 0x7FFFFFFF);
}
__device__ __forceinline__ float fdec(int i) {
  return __int_as_float(i >= 0 ? i : (i ^ 0x7FFFFFFF));
}

// ---------------------------------------------------------------------------
// C[M,N] = A[M,K] @ B[K,N], fp32-exact via V_WMMA_F32_16X16X4_F32.
// One wave32 per 16x16 output tile; K consumed 4 at a time.
// A 16x4 layout: lanes 0-15 hold K={k,k+1}, lanes 16-31 hold K={k+2,k+3}.
// C/D layout: VGPR i -> row m0+i (lanes 0-15) / m0+i+8 (lanes 16-31).
// ---------------------------------------------------------------------------
__global__ void gat_gemm_wmma(const float* __restrict__ A,
                              const float* __restrict__ B,
                              float* __restrict__ C,
                              int M, int K, int N) {
  const int lane  = threadIdx.x & 31;
  const int wave  = threadIdx.x >> 5;
  const int mtile = blockIdx.x * (blockDim.x >> 5) + wave;
  if (mtile * 16 >= M) return;           // wave-uniform: EXEC stays all-1s
  const int ntile = blockIdx.y;
  const int half  = lane >> 4;           // 0 -> K pair {k,k+1}; 1 -> {k+2,k+3}
  const int l16   = lane & 15;
  const int m0 = mtile * 16;
  const int n0 = ntile * 16;
  const float* arow = A + (size_t)(m0 + l16) * K;
  v8f acc = {};
  for (int k = 0; k < K; k += 4) {
    const int ka = k + 2 * half;
    v2f a, b;
    a.x = arow[ka];
    a.y = arow[ka + 1];
    b.x = B[(size_t)ka * N + n0 + l16];
    b.y = B[(size_t)(ka + 1) * N + n0 + l16];
    acc = __builtin_amdgcn_wmma_f32_16x16x4_f32(
        false, a, false, b, (short)0, acc, false, false);
  }
#pragma unroll
  for (int i = 0; i < 8; ++i) {
    C[(size_t)(m0 + i + 8 * half) * N + n0 + l16] = acc[i];
  }
}

// Per-(node, head) attention logit halves: asrc[n,h] = <h[n,h,:], a_src[h,:]>
__global__ void alpha_k(const float* __restrict__ h,
                        const float* __restrict__ a_src,
                        const float* __restrict__ a_dst,
                        float* __restrict__ asrc, float* __restrict__ adst,
                        int Nn, int H, int C) {
  int i = blockIdx.x * blockDim.x + threadIdx.x;
  if (i >= Nn * H) return;
  int n = i / H, hh = i % H;
  const float* hp = h + (size_t)n * H * C + (size_t)hh * C;
  float s = 0.f, d = 0.f;
  for (int c = 0; c < C; ++c) {
    float v = hp[c];
    s += v * a_src[hh * C + c];
    d += v * a_dst[hh * C + c];
  }
  asrc[i] = s;
  adst[i] = d;
}

__global__ void fill_i32(int* __restrict__ p, int v, int n) {
  int i = blockIdx.x * blockDim.x + threadIdx.x;
  if (i < n) p[i] = v;
}

// Edge logit: leaky_relu(asrc[src] + adst[dst]); edges e >= E are self-loops.
__device__ __forceinline__ void edge_sd(const int* __restrict__ srcv,
                                        const int* __restrict__ dstv,
                                        int e, int E, int& s, int& d) {
  if (e < E) { s = srcv[e]; d = dstv[e]; }
  else       { s = e - E;  d = e - E; }
}

__global__ void edge_max_k(const int* __restrict__ srcv, const int* __restrict__ dstv,
                           int E, int Nn, int H,
                           const float* __restrict__ asrc, const float* __restrict__ adst,
                           int* __restrict__ mmax) {
  int i = blockIdx.x * blockDim.x + threadIdx.x;
  if (i >= (E + Nn) * H) return;
  int e = i / H, hh = i % H;
  int s, d; edge_sd(srcv, dstv, e, E, s, d);
  float x = asrc[s * H + hh] + adst[d * H + hh];
  x = x > 0.f ? x : NEG_SLOPE * x;
  atomicMax(&mmax[d * H + hh], fenc(x));
}

__global__ void edge_sum_k(const int* __restrict__ srcv, const int* __restrict__ dstv,
                           int E, int Nn, int H,
                           const float* __restrict__ asrc, const float* __restrict__ adst,
                           const int* __restrict__ mmax, float* __restrict__ denom) {
  int i = blockIdx.x * blockDim.x + threadIdx.x;
  if (i >= (E + Nn) * H) return;
  int e = i / H, hh = i % H;
  int s, d; edge_sd(srcv, dstv, e, E, s, d);
  float x = asrc[s * H + hh] + adst[d * H + hh];
  x = x > 0.f ? x : NEG_SLOPE * x;
  float ex = expf(x - fdec(mmax[d * H + hh]));
  atomicAdd(&denom[d * H + hh], ex);
}

// Weighted scatter-add: out[dst,h,:] += alpha * h[src,h,:]
__global__ void edge_agg_k(const int* __restrict__ srcv, const int* __restrict__ dstv,
                           int E, int Nn, int H, int C,
                           const float* __restrict__ asrc, const float* __restrict__ adst,
                           const int* __restrict__ mmax, const float* __restrict__ denom,
                           const float* __restrict__ h, float* __restrict__ out) {
  int i = blockIdx.x * blockDim.x + threadIdx.x;
  if (i >= (E + Nn) * H) return;
  int e = i / H, hh = i % H;
  int s, d; edge_sd(srcv, dstv, e, E, s, d);
  float x = asrc[s * H + hh] + adst[d * H + hh];
  x = x > 0.f ? x : NEG_SLOPE * x;
  float ex = expf(x - fdec(mmax[d * H + hh]));
  float alpha = ex / (denom[d * H + hh] + 1e-16f);
  const float* hp = h + (size_t)s * H * C + (size_t)hh * C;
  float* op = out + (size_t)d * H * C + (size_t)hh * C;
  __builtin_prefetch(hp, 0, 1);   // global_prefetch_b8 on the gathered row
  for (int c = 0; c < C; ++c) {
    atomicAdd(&op[c], alpha * hp[c]);
  }
}

__global__ void bias_elu_k(float* __restrict__ h, const float* __restrict__ b,
                           int total, int F) {
  int i = blockIdx.x * blockDim.x + threadIdx.x;
  if (i >= total) return;
  float v = h[i] + b[i % F];
  h[i] = v > 0.f ? v : (expf(v) - 1.0f);
}

__global__ void final_linear_k(const float* __restrict__ h, const float* __restrict__ W,
                               const float* __restrict__ b, float* __restrict__ out,
                               int Nn, int K, int N) {
  int i = blockIdx.x * blockDim.x + threadIdx.x;
  if (i >= Nn * N) return;
  int n = i / N, j = i % N;
  float s = b[j];
  const float* hp = h + (size_t)n * K;
  for (int k = 0; k < K; ++k) s += hp[k] * W[k * N + j];
  out[i] = s;
}

// ---------------------------------------------------------------------------
static void gat_layer(const float* hin, const float* W, const float* avs,
                      const float* avd, const float* bias,
                      float* hmid, float* hout,
                      float* asrcb, float* adstb, int* mmax, float* denom,
                      const int* srcv, const int* dstv,
                      int E, int Nn, int Kdim, int H, int C, hipStream_t stream) {
  const int F = H * C;
  // 1) h = hin @ W  (WMMA)
  dim3 gemm_grid((Nn / 16 + 7) / 8, F / 16);
  gat_gemm_wmma<<<gemm_grid, 256, 0, stream>>>(hin, W, hmid, Nn, Kdim, F);
  // 2) per-node attention logit halves
  const int nh = Nn * H;
  alpha_k<<<(nh + 255) / 256, 256, 0, stream>>>(hmid, avs, avd, asrcb, adstb, Nn, H, C);
  // 3) segment softmax: max, sum, then weighted scatter-add
  fill_i32<<<(nh + 255) / 256, 256, 0, stream>>>(mmax, (int)0x80000000, nh);
  hipMemsetAsync(denom, 0, (size_t)nh * sizeof(float), stream);
  const int tote = (E + Nn) * H;
  edge_max_k<<<(tote + 255) / 256, 256, 0, stream>>>(srcv, dstv, E, Nn, H, asrcb, adstb, mmax);
  edge_sum_k<<<(tote + 255) / 256, 256, 0, stream>>>(srcv, dstv, E, Nn, H, asrcb, adstb, mmax, denom);
  hipMemsetAsync(hout, 0, (size_t)Nn * F * sizeof(float), stream);
  edge_agg_k<<<(tote + 255) / 256, 256, 0, stream>>>(srcv, dstv, E, Nn, H, C,
                                                    asrcb, adstb, mmax, denom, hmid, hout);
  // 4) bias + ELU (in place)
  const int nf = Nn * F;
  bias_elu_k<<<(nf + 255) / 256, 256, 0, stream>>>(hout, bias, nf, F);
}

extern "C" void kernel_launch(void* const* d_in, const int* in_sizes, int n_in,
                              void* d_out, int out_size, void* d_ws, size_t ws_size,
                              hipStream_t stream) {
  const float* x    = (const float*)d_in[0];
  const int*   ei   = (const int*)d_in[1];
  const float* W1   = (const float*)d_in[2];
  const float* as1  = (const float*)d_in[3];
  const float* ad1  = (const float*)d_in[4];
  const float* b1   = (const float*)d_in[5];
  const float* W2   = (const float*)d_in[6];
  const float* as2  = (const float*)d_in[7];
  const float* ad2  = (const float*)d_in[8];
  const float* b2   = (const float*)d_in[9];
  const float* W3   = (const float*)d_in[10];
  const float* as3  = (const float*)d_in[11];
  const float* ad3  = (const float*)d_in[12];
  const float* b3   = (const float*)d_in[13];
  const float* Wout = (const float*)d_in[14];
  const float* bout = (const float*)d_in[15];

  const int Nn = in_sizes[0] / 128;   // 100000
  const int E  = in_sizes[1] / 2;     // 1600000
  const int* srcv = ei;
  const int* dstv = ei + E;

  // Workspace layout (~115 MB): two ping-pong node buffers + per-(node,head) scratch.
  float* hA    = (float*)d_ws;                 // Nn*128
  float* hB    = hA  + (size_t)Nn * 128;       // Nn*128
  float* asrcb = hB  + (size_t)Nn * 128;       // Nn*8
  float* adstb = asrcb + (size_t)Nn * 8;       // Nn*8
  int*   mmax  = (int*)(adstb + (size_t)Nn * 8);   // Nn*8
  float* denom = (float*)(mmax + (size_t)Nn * 8);  // Nn*8

  // Layer 1: 128 -> 8 x 16 (concat)
  gat_layer(x,  W1, as1, ad1, b1, hA, hB, asrcb, adstb, mmax, denom,
            srcv, dstv, E, Nn, 128, 8, 16, stream);
  // Layer 2: 128 -> 8 x 16 (concat)   [reads hB, writes hA, scatters back into hB]
  gat_layer(hB, W2, as2, ad2, b2, hA, hB, asrcb, adstb, mmax, denom,
            srcv, dstv, E, Nn, 128, 8, 16, stream);
  // Layer 3: 128 -> 1 x 64 (mean over a single head == identity)
  gat_layer(hB, W3, as3, ad3, b3, hA, hB, asrcb, adstb, mmax, denom,
            srcv, dstv, E, Nn, 128, 1, 64, stream);
  // Output: [Nn,64] @ [64,2] + bout
  final_linear_k<<<(Nn * 2 + 255) / 256, 256, 0, stream>>>(
      hB, Wout, bout, (float*)d_out, Nn, 64, 2);
}